// NopadBaichuanAttention_21096879358033
// MI455X (gfx1250) — compile-verified
//
#include <hip/hip_runtime.h>
#include <hip/hip_bf16.h>

// Problem constants (from reference)
#define BQ   8        // decode batch
#define HID  4096     // hidden (= K = N for all GEMMs here)
#define NH   32       // heads
#define HD   128      // head dim
#define BS   16       // cache block size
#define MB   128      // max blocks per seq

typedef float v2f __attribute__((ext_vector_type(2)));
typedef float v8f __attribute__((ext_vector_type(8)));

// ---------------------------------------------------------------------------
// Pad kernel: build zero-padded 16-row A matrices so the GEMM loop has no
// predication. A16q rows 0-7 = hidden, rows 8-15 = 0. A16o rows 8-15 = 0
// (rows 0-7 are written later by the attention kernel).
// ---------------------------------------------------------------------------
__global__ __launch_bounds__(256) void pad_kernel(
    const float* __restrict__ hidden,  // [8, HID]
    float* __restrict__ A16q,          // [16, HID]
    float* __restrict__ A16o)          // [16, HID]
{
    const int i = blockIdx.x * blockDim.x + threadIdx.x;  // over 16*HID
    const int r = i / HID;
    A16q[i] = (r < BQ) ? hidden[i] : 0.f;
    if (r >= BQ) A16o[i] = 0.f;
}

// ---------------------------------------------------------------------------
// Skinny GEMM: Out[t][m][n] = sum_k A16[m][k] * W[t][k][n], valid m in [0,8).
// A16 is zero-padded to 16 rows -> unconditional loads. K = N = HID are
// compile-time so all B loads use immediate offsets off one address pair.
// One wave per 16-wide N tile; fp32 WMMA 16x16x4 streams K.
// grid.x = T * (HID/16), block = 32 (EXEC all ones, as WMMA requires).
// ---------------------------------------------------------------------------
__global__ __launch_bounds__(32) void wmma_gemm16(
    const float* __restrict__ A16,  // [16, HID], rows 8-15 zero
    const float* __restrict__ W,    // [T, HID, HID]
    float* __restrict__ Out)        // [T, 8, HID]
{
    const int bid = blockIdx.x;
    const int t   = bid >> 8;              // / (HID/16)
    const int n0  = (bid & 255) * 16;      // % (HID/16) * 16

    const int lane = threadIdx.x;
    const int half = lane >> 4;            // K sub-pair selector
    const int lm   = lane & 15;            // A: row m; B/D: column n

    // A fragment: lane holds A16[lm][k + 2*half + {0,1}] -> aligned b64 load
    const float* __restrict__ aptr = A16 + (size_t)lm * HID + 2 * half;
    // B fragment: VGPR0 = W[k+2*half][n0+lm], VGPR1 = W[k+2*half+1][n0+lm]
    const float* __restrict__ bptr =
        W + (size_t)t * HID * HID + (size_t)(2 * half) * HID + n0 + lm;
    float* __restrict__ Ot = Out + (size_t)t * 8 * HID;

    v8f acc = {0.f, 0.f, 0.f, 0.f, 0.f, 0.f, 0.f, 0.f};

    #pragma unroll 8
    for (int k = 0; k < HID; k += 4) {
        v2f a = *(const v2f*)aptr;
        v2f b;
        b.x = bptr[0];
        b.y = bptr[HID];
        if ((k & 31) == 0)  // one prefetch per unrolled block (64 rows ahead)
            __builtin_prefetch(bptr + (size_t)64 * HID, 0, 1);
        acc = __builtin_amdgcn_wmma_f32_16x16x4_f32(
            /*neg_a=*/false, a, /*neg_b=*/false, b,
            /*c_mod=*/(short)0, acc, /*reuse_a=*/false, /*reuse_b=*/false);
        aptr += 4;
        bptr += (size_t)4 * HID;
    }

    // D layout: VGPR r -> row r (lanes 0-15); rows 8-15 (lanes 16-31) invalid
    if (half == 0) {
        #pragma unroll
        for (int r = 0; r < 8; ++r)
            Ot[(size_t)r * HID + n0 + lm] = acc[r];
    }
}

// ---------------------------------------------------------------------------
// RoPE on the new token's q and k: out = x*cos + rotate_half(x)*sin
// qk layout: [2, B, HID] (t=0 q, t=1 k); cos/sin: [B, HD]
// ---------------------------------------------------------------------------
__global__ __launch_bounds__(256) void rope_kernel(
    const float* __restrict__ qk, const float* __restrict__ cosb,
    const float* __restrict__ sinb, float* __restrict__ out, int total)
{
    const int i = blockIdx.x * blockDim.x + threadIdx.x;
    if (i >= total) return;
    const int b = (i / HID) % BQ;
    const int p = i & (HD - 1);
    const float x = qk[i];
    const float partner = (p < HD / 2) ? -qk[i + HD / 2] : qk[i - HD / 2];
    out[i] = x * cosb[b * HD + p] + partner * sinb[b * HD + p];
}

// ---------------------------------------------------------------------------
// Flash-decoding attention over the paged KV cache.
// One block per (b, head); 8 waves split kv positions; LDS combine.
// Output goes into the padded [16, HID] buffer (rows 0-7).
// ---------------------------------------------------------------------------
__global__ __launch_bounds__(256) void attn_kernel(
    const float* __restrict__ qr,    // [B, HID] roped q
    const float* __restrict__ kr,    // [B, HID] roped k (new token)
    const float* __restrict__ vnew,  // [B, HID] new v
    const float* __restrict__ kc,    // [NB, NH, BS, HD]
    const float* __restrict__ vc,    // [NB, NH, BS, HD]
    const int*   __restrict__ btab,  // [B, MB]
    const int*   __restrict__ slen,  // [B]
    float* __restrict__ out)         // [16, HID] padded; rows 0-7 written
{
    __shared__ float q_s[HD];
    __shared__ float m_s[8], l_s[8];
    __shared__ float acc_s[8][HD];

    const int bh   = blockIdx.x;
    const int b    = bh >> 5;       // / NH
    const int h    = bh & 31;       // % NH
    const int tid  = threadIdx.x;
    const int w    = tid >> 5;      // wave id 0..7
    const int lane = tid & 31;

    const float scale = 0.08838834764831845f;  // 1/sqrt(128)
    if (tid < HD)
        q_s[tid] = qr[(size_t)b * HID + h * HD + tid] * scale;
    __syncthreads();

    const float4 q4 = *(const float4*)&q_s[lane * 4];
    const int seq = slen[b];

    float  m = -3.0e38f, l = 0.f;
    float4 a4 = {0.f, 0.f, 0.f, 0.f};

    for (int pos = w; pos < seq; pos += 8) {
        const float* kptr;
        const float* vptr;
        if (pos == seq - 1) {                       // new token overrides cache
            kptr = kr   + (size_t)b * HID + h * HD;
            vptr = vnew + (size_t)b * HID + h * HD;
        } else {
            const int blk = btab[b * MB + (pos >> 4)];
            const size_t base = (((size_t)blk * NH + h) * BS + (pos & (BS - 1))) * HD;
            kptr = kc + base;
            vptr = vc + base;
        }
        const float4 k4 = *(const float4*)&kptr[lane * 4];
        float s = q4.x * k4.x + q4.y * k4.y + q4.z * k4.z + q4.w * k4.w;
        s += __shfl_xor(s, 16, 32);
        s += __shfl_xor(s,  8, 32);
        s += __shfl_xor(s,  4, 32);
        s += __shfl_xor(s,  2, 32);
        s += __shfl_xor(s,  1, 32);

        const float  mn = fmaxf(m, s);
        const float  so = __expf(m - mn);
        const float  p  = __expf(s - mn);
        const float4 v4 = *(const float4*)&vptr[lane * 4];
        a4.x = a4.x * so + p * v4.x;
        a4.y = a4.y * so + p * v4.y;
        a4.z = a4.z * so + p * v4.z;
        a4.w = a4.w * so + p * v4.w;
        l = l * so + p;
        m = mn;
    }

    if (lane == 0) { m_s[w] = m; l_s[w] = l; }
    *(float4*)&acc_s[w][lane * 4] = a4;
    __syncthreads();

    if (tid < HD) {
        float mt = m_s[0];
        #pragma unroll
        for (int i = 1; i < 8; ++i) mt = fmaxf(mt, m_s[i]);
        float L = 0.f, num = 0.f;
        #pragma unroll
        for (int i = 0; i < 8; ++i) {
            const float e = __expf(m_s[i] - mt);
            L   += l_s[i] * e;
            num += acc_s[i][tid] * e;
        }
        out[(size_t)b * HID + h * HD + tid] = num / L;
    }
}

// ---------------------------------------------------------------------------
extern "C" void kernel_launch(void* const* d_in, const int* in_sizes, int n_in,
                              void* d_out, int out_size, void* d_ws, size_t ws_size,
                              hipStream_t stream) {
    const float* hidden = (const float*)d_in[0];   // [8, 4096]
    const float* qkvw   = (const float*)d_in[1];   // [3, 4096, 4096]
    const float* ow     = (const float*)d_in[2];   // [4096, 4096]
    const float* kc     = (const float*)d_in[3];   // [NB, NH, BS, HD]
    const float* vc     = (const float*)d_in[4];
    const float* cosb   = (const float*)d_in[5];   // [8, 128]
    const float* sinb   = (const float*)d_in[6];
    const int*   btab   = (const int*)d_in[7];     // [8, 128]
    const int*   slen   = (const int*)d_in[8];     // [8]

    float* ws       = (float*)d_ws;
    float* A16q     = ws;                          // 16*4096 (padded hidden)
    float* A16o     = A16q + 16 * HID;             // 16*4096 (padded attn out)
    float* qkv_raw  = A16o + 16 * HID;             // 3*8*4096
    float* qk_rope  = qkv_raw + 3 * BQ * HID;      // 2*8*4096 (roped q,k)

    // 0) zero-pad A matrices (A16o rows 8-15 zeroed every call)
    pad_kernel<<<(16 * HID) / 256, 256, 0, stream>>>(hidden, A16q, A16o);
    // 1) fused QKV projection (WMMA fp32): 3*(4096/16) = 768 one-wave blocks
    wmma_gemm16<<<3 * (HID / 16), 32, 0, stream>>>(A16q, qkvw, qkv_raw);
    // 2) RoPE on q,k of the new token
    rope_kernel<<<(2 * BQ * HID) / 256, 256, 0, stream>>>(
        qkv_raw, cosb, sinb, qk_rope, 2 * BQ * HID);
    // 3) flash-decoding attention over paged cache -> padded A16o rows 0-7
    attn_kernel<<<BQ * NH, 256, 0, stream>>>(
        qk_rope,                  // roped q
        qk_rope + BQ * HID,       // roped k
        qkv_raw + 2 * BQ * HID,   // v (no rope)
        kc, vc, btab, slen, A16o);
    // 4) output projection (WMMA fp32) straight into d_out
    wmma_gemm16<<<HID / 16, 32, 0, stream>>>(A16o, ow, (float*)d_out);
}